// Raindrop_36601711296513
// MI455X (gfx1250) — compile-verified
//
#include <hip/hip_runtime.h>
#include <math.h>

// D = A(16x4 f32) x B(4x16 f32) + C(16x16 f32), wave32 WMMA fragments
typedef __attribute__((ext_vector_type(2))) float v2f;
typedef __attribute__((ext_vector_type(4))) float f4;
typedef __attribute__((ext_vector_type(8))) float v8f;

namespace {
constexpr int S = 64, OBS = 4, EMB = 32, PE = 16, ATTN = 10;
constexpr int KP = 32;        // receiver feature dim 26 padded to 32 (zero cols)
constexpr int THREADS = 128;  // 4 x wave32
}

__global__ __launch_bounds__(THREADS) void raindrop_fused(
    const float* __restrict__ x,       // [BT, S, OBS]
    const float* __restrict__ times,   // [BT]
    const float* __restrict__ mask,    // [BT, S]
    const float* __restrict__ obs_w,   // [S, OBS, EMB]
    const float* __restrict__ attn_w,  // [S, ATTN]
    const float* __restrict__ recv_W,  // [EMB, ATTN+PE]
    const float* __restrict__ recv_b,  // [ATTN+PE]
    float* __restrict__ out)           // [BT, S, S]
{
    // Phase 1-3 scratch is dead before the alpha tile is materialized, so the
    // 16 KB output staging buffer aliases it.  Total LDS ~= 32 KB.
    __shared__ __align__(16) union SharedU {
        struct {
            float x[S * OBS];    // 1 KB
            float mask[S];
            float pe[PE];
            float W[EMB * KP];   // recv_W zero-padded to 32 cols
            float b[KP];
            float h[S * EMB];    // 8 KB
        } a;
        float alpha[S * S];      // 16 KB relu'd output tile
    } u;
    __shared__ float s_hq[S * KP];     // 8 KB
    __shared__ float s_keyT[KP * S];   // 8 KB, keyT[k][u]

    const int bt   = blockIdx.x;
    const int tid  = threadIdx.x;
    const int lane = tid & 31;
    const int wave = tid >> 5;
    const int half = lane >> 4;   // 0: lanes 0-15, 1: lanes 16-31
    const int l16  = lane & 15;

    // ---- phase 1: stage operands into LDS ----
    // x / mask are streamed once across the whole grid -> non-temporal loads
    // so they do not evict the hot obs_w working set from L2.
    for (int i = tid; i < S * OBS; i += THREADS)
        u.a.x[i] = __builtin_nontemporal_load(&x[(size_t)bt * (S * OBS) + i]);
    for (int i = tid; i < S; i += THREADS)
        u.a.mask[i] = __builtin_nontemporal_load(&mask[(size_t)bt * S + i]);
    for (int i = tid; i < EMB * KP; i += THREADS) {
        int e = i >> 5, d = i & (KP - 1);
        u.a.W[i] = (d < ATTN + PE) ? recv_W[e * (ATTN + PE) + d] : 0.0f;
    }
    if (tid < KP)
        u.a.b[tid] = (tid < ATTN + PE) ? recv_b[tid] : 0.0f;
    if (tid < PE / 2) {
        // div[k] = 10000^(-k/8); pe = interleave(sin, cos)
        float tv  = times[bt];
        float dv  = expf((float)tid * (-9.210340371976184f / 8.0f));
        float ang = tv * dv;
        u.a.pe[2 * tid]     = sinf(ang);
        u.a.pe[2 * tid + 1] = cosf(ang);
    }
    __syncthreads();

    // ---- phase 2: h = relu(x @ obs_w) * mask ; keyT = [attn | pe | 0]^T ----
    for (int i = tid; i < S * EMB; i += THREADS) {
        int s = i >> 5, e = i & (EMB - 1);
        float acc = 0.0f;
        #pragma unroll
        for (int o = 0; o < OBS; ++o)
            acc = fmaf(u.a.x[s * OBS + o], obs_w[s * (OBS * EMB) + o * EMB + e], acc);
        u.a.h[i] = fmaxf(acc, 0.0f) * u.a.mask[s];
    }
    for (int i = tid; i < KP * S; i += THREADS) {
        int k = i >> 6, uu = i & (S - 1);
        float v = 0.0f;
        if (k < ATTN)           v = attn_w[uu * ATTN + k];
        else if (k < ATTN + PE) v = u.a.pe[k - ATTN];
        s_keyT[i] = v;
    }
    __syncthreads();

    // ---- phase 3: hq = h @ recv_W + recv_b  (v_wmma_f32_16x16x4_f32) ----
    {
        const int s0 = wave * 16;
        #pragma unroll
        for (int n = 0; n < 2; ++n) {
            const int n0 = n * 16;
            const float bval = u.a.b[n0 + l16];
            v8f c;
            #pragma unroll
            for (int r = 0; r < 8; ++r) c[r] = bval;   // seed C with bias
            #pragma unroll
            for (int k0 = 0; k0 < EMB; k0 += 4) {
                const int kk = k0 + 2 * half;
                v2f a, bb;
                a.x  = u.a.h[(s0 + l16) * EMB + kk];
                a.y  = u.a.h[(s0 + l16) * EMB + kk + 1];
                bb.x = u.a.W[kk * KP + n0 + l16];
                bb.y = u.a.W[(kk + 1) * KP + n0 + l16];
                c = __builtin_amdgcn_wmma_f32_16x16x4_f32(
                        false, a, false, bb, (short)0, c, false, false);
            }
            #pragma unroll
            for (int r = 0; r < 8; ++r)
                s_hq[(s0 + r + 8 * half) * KP + n0 + l16] = c[r];
        }
    }
    __syncthreads();   // after this, u.a.* is dead -> u.alpha may be written

    // ---- phase 4: alpha = relu(hq @ keyT) -> LDS tile ----
    {
        const int s0 = wave * 16;
        #pragma unroll
        for (int n = 0; n < 4; ++n) {
            const int u0 = n * 16;
            v8f c;
            #pragma unroll
            for (int r = 0; r < 8; ++r) c[r] = 0.0f;
            #pragma unroll
            for (int k0 = 0; k0 < KP; k0 += 4) {
                const int kk = k0 + 2 * half;
                v2f a, bb;
                a.x  = s_hq[(s0 + l16) * KP + kk];
                a.y  = s_hq[(s0 + l16) * KP + kk + 1];
                bb.x = s_keyT[kk * S + u0 + l16];
                bb.y = s_keyT[(kk + 1) * S + u0 + l16];
                c = __builtin_amdgcn_wmma_f32_16x16x4_f32(
                        false, a, false, bb, (short)0, c, false, false);
            }
            #pragma unroll
            for (int r = 0; r < 8; ++r)
                u.alpha[(s0 + r + 8 * half) * S + u0 + l16] = fmaxf(c[r], 0.0f);
        }
    }
    __syncthreads();

    // ---- phase 5: drain tile with coalesced non-temporal b128 stores ----
    {
        const f4* src = (const f4*)u.alpha;
        f4* dst = (f4*)(out + (size_t)bt * (S * S));
        #pragma unroll
        for (int i = tid; i < (S * S) / 4; i += THREADS)
            __builtin_nontemporal_store(src[i], &dst[i]);
    }
}

extern "C" void kernel_launch(void* const* d_in, const int* in_sizes, int n_in,
                              void* d_out, int out_size, void* d_ws, size_t ws_size,
                              hipStream_t stream) {
    const float* x      = (const float*)d_in[0];
    const float* times  = (const float*)d_in[1];
    const float* mask   = (const float*)d_in[2];
    const float* obs_w  = (const float*)d_in[3];
    const float* attn_w = (const float*)d_in[4];
    const float* recv_W = (const float*)d_in[5];
    const float* recv_b = (const float*)d_in[6];
    float* out = (float*)d_out;

    const int BT = in_sizes[1];  // B*T = 16384 (times is [B,T])
    dim3 grid(BT), block(THREADS);
    raindrop_fused<<<grid, block, 0, stream>>>(x, times, mask, obs_w, attn_w,
                                               recv_W, recv_b, out);
}